// MyModel_90254442758417
// MI455X (gfx1250) — compile-verified
//
#include <hip/hip_runtime.h>
#include <hip/hip_bf16.h>

// Sizes from the reference
#define BB   128
#define SS   512
#define EE   768
#define HH   16
#define SNN  100
#define LMAXX 5
#define DHH  48        // E/H
#define E3   2304      // 3*E

typedef float v2f __attribute__((ext_vector_type(2)));
typedef float v8f __attribute__((ext_vector_type(8)));

// fp32 WMMA: D(16x16) = A(16x4) * B(4x16) + C, full fp32 precision.
// Fragment layouts per CDNA5 ISA 7.12.2:
//   A: lane L holds rows m=L%16; vgpr v holds k = 2*(L/16) + v
//   B: symmetric, lane L holds col n=L%16; vgpr v holds k = 2*(L/16) + v
//   D: vgpr r holds row m = r + 8*(L/16), col n = L%16
__device__ __forceinline__ v8f wmma_f32(v2f a, v2f b, v8f c) {
    return __builtin_amdgcn_wmma_f32_16x16x4_f32(
        /*neg_a=*/false, a, /*neg_b=*/false, b,
        /*c_mod=*/(short)0, c, /*reuse_a=*/false, /*reuse_b=*/false);
}

// ---------------------------------------------------------------------------
// Kernel 1: masked gather + mean pooling.  pun[b, :] = mean over valid rows.
// ---------------------------------------------------------------------------
__global__ void pool_kernel(const float* __restrict__ lhs,
                            const int* __restrict__ loc,
                            float* __restrict__ pun) {
    const int b = blockIdx.x;
    int l[LMAXX];
    int cnt = 0;
    #pragma unroll
    for (int i = 0; i < LMAXX; ++i) {
        l[i] = loc[b * LMAXX + i];
        if (l[i] != -1) cnt++;
    }
    const float inv = 1.0f / (float)cnt;
    for (int j = threadIdx.x; j < EE; j += blockDim.x) {
        float s = 0.f;
        #pragma unroll
        for (int i = 0; i < LMAXX; ++i) {
            if (l[i] != -1) s += lhs[((size_t)b * SS + l[i]) * EE + j];
        }
        pun[b * EE + j] = s * inv;
    }
}

// ---------------------------------------------------------------------------
// Kernel 2: generic fp32 WMMA GEMM, C(MxN) = A(Mx768) @ W(Nx768)^T + bias
// A row-major (stride 768), W row-major (stride 768), C row-major (stride N).
// grid = (M/16, ceil(N/16/8)), block = 256 (8 waves, 1 n-tile per wave).
// A 16x768 tile staged in LDS (48 KB), shared by all 8 waves.
// ---------------------------------------------------------------------------
__global__ void gemm768_wmma(const float* __restrict__ A,
                             const float* __restrict__ W,
                             const float* __restrict__ bias,
                             float* __restrict__ C, int N) {
    __shared__ float As[16 * 768];  // 48 KB
    const int tid = threadIdx.x;
    const int mt  = blockIdx.x;

    // Cooperative stage of the 16x768 A tile.
    const float* Arow = A + (size_t)mt * 16 * 768;
    for (int i = tid; i < 16 * 768; i += 256) As[i] = Arow[i];
    __syncthreads();

    const int w  = tid >> 5;          // wave id 0..7
    const int L  = tid & 31;          // lane id
    const int nt = blockIdx.y * 8 + w;
    const int n0 = nt * 16;
    if (n0 >= N) return;              // whole wave exits together (EXEC stays full)

    const int mrow  = L & 15;
    const int khalf = (L >> 4) << 1;  // 0 or 2
    const int ncol  = n0 + mrow;
    const int nsafe = ncol < N ? ncol : (N - 1);   // clamp keeps EXEC all-1s;
                                                   // garbage cols are never stored
    const float* wrow = W + (size_t)nsafe * 768;
    const float* arow = As + mrow * 768;

    v8f acc = {};
    #pragma unroll 8
    for (int kk = 0; kk < 768; kk += 4) {
        v2f a, b;
        const int kc = kk + khalf;
        a.x = arow[kc];
        a.y = arow[kc + 1];
        b.x = wrow[kc];
        b.y = wrow[kc + 1];
        acc = wmma_f32(a, b, acc);
    }

    const float bv   = bias[nsafe];
    const int mbase  = mt * 16 + ((L >> 4) << 3);
    if (ncol < N) {
        #pragma unroll
        for (int r = 0; r < 8; ++r)
            C[(size_t)(mbase + r) * N + ncol] = acc[r] + bv;
    }
}

// ---------------------------------------------------------------------------
// Kernel 3: per-head attention over the batch dim (only the n=0 slice of the
// reference survives).  proj holds [q | k | v] as (128 x 2304).
// One block per head; wave w owns row strip m0 = 16*w.
//   scores(16x128) = (q_h*scale) @ k_h^T     -> LDS
//   row softmax (wave32 shuffle reductions)
//   ctx(16x48)    = attn(16x128) @ v_h(128x48)
// ---------------------------------------------------------------------------
__global__ void attn_kernel(const float* __restrict__ proj,
                            float* __restrict__ ctx) {
    __shared__ float sS[128 * 128];   // 64 KB score tile (wave-private strips)
    const int h   = blockIdx.x;
    const int tid = threadIdx.x;
    const int w   = tid >> 5;
    const int L   = tid & 31;
    const int m0  = w * 16;
    const int mrow  = L & 15;
    const int khalf = (L >> 4) << 1;
    const int rbase = (L >> 4) << 3;
    const float scale = 0.14433756729740643f;  // 1/sqrt(48)

    const float* qp = proj + h * DHH;            // q columns
    const float* kp = proj + EE + h * DHH;       // k columns
    const float* vp = proj + 2 * EE + h * DHH;   // v columns

    // ---- scores: 8 n-tiles of 16, K = 48 ----
    for (int nt = 0; nt < 8; ++nt) {
        v8f acc = {};
        #pragma unroll
        for (int kk = 0; kk < DHH; kk += 4) {
            const int kc = kk + khalf;
            v2f a, b;
            a.x = qp[(size_t)(m0 + mrow) * E3 + kc] * scale;
            a.y = qp[(size_t)(m0 + mrow) * E3 + kc + 1] * scale;
            b.x = kp[(size_t)(nt * 16 + mrow) * E3 + kc];
            b.y = kp[(size_t)(nt * 16 + mrow) * E3 + kc + 1];
            acc = wmma_f32(a, b, acc);
        }
        const int col = nt * 16 + mrow;
        #pragma unroll
        for (int r = 0; r < 8; ++r)
            sS[(m0 + rbase + r) * 128 + col] = acc[r];
    }

    // ---- softmax over each of this wave's 16 rows (128 cols, 4 per lane) ----
    for (int r = 0; r < 16; ++r) {
        float* row = sS + (m0 + r) * 128;
        float x0 = row[L], x1 = row[L + 32], x2 = row[L + 64], x3 = row[L + 96];
        float mx = fmaxf(fmaxf(x0, x1), fmaxf(x2, x3));
        #pragma unroll
        for (int off = 16; off > 0; off >>= 1)
            mx = fmaxf(mx, __shfl_xor(mx, off, 32));
        x0 = __expf(x0 - mx); x1 = __expf(x1 - mx);
        x2 = __expf(x2 - mx); x3 = __expf(x3 - mx);
        float sm = x0 + x1 + x2 + x3;
        #pragma unroll
        for (int off = 16; off > 0; off >>= 1)
            sm += __shfl_xor(sm, off, 32);
        const float inv = 1.0f / sm;
        row[L] = x0 * inv; row[L + 32] = x1 * inv;
        row[L + 64] = x2 * inv; row[L + 96] = x3 * inv;
    }

    // ---- ctx: (16 x 48) = attn(16x128) @ v(128x48), 3 n-tiles, K = 128 ----
    for (int nt = 0; nt < 3; ++nt) {
        v8f acc = {};
        #pragma unroll 8
        for (int kk = 0; kk < 128; kk += 4) {
            const int kc = kk + khalf;
            v2f a, b;
            a.x = sS[(m0 + mrow) * 128 + kc];
            a.y = sS[(m0 + mrow) * 128 + kc + 1];
            b.x = vp[(size_t)kc * E3 + nt * 16 + mrow];
            b.y = vp[(size_t)(kc + 1) * E3 + nt * 16 + mrow];
            acc = wmma_f32(a, b, acc);
        }
        const int ncol = h * DHH + nt * 16 + mrow;
        #pragma unroll
        for (int r = 0; r < 8; ++r)
            ctx[(size_t)(m0 + rbase + r) * EE + ncol] = acc[r];
    }
}

// ---------------------------------------------------------------------------
extern "C" void kernel_launch(void* const* d_in, const int* in_sizes, int n_in,
                              void* d_out, int out_size, void* d_ws, size_t ws_size,
                              hipStream_t stream) {
    const float* lhs    = (const float*)d_in[0];  // (128, 512, 768)
    // d_in[1] = sense_emb: provably dead w.r.t. the output -> never read
    const int*   loc    = (const int*)  d_in[2];  // (128, 5)
    const float* w_in   = (const float*)d_in[3];  // (2304, 768)
    const float* b_in   = (const float*)d_in[4];  // (2304,)
    const float* w_out  = (const float*)d_in[5];  // (768, 768)
    const float* b_out  = (const float*)d_in[6];  // (768,)
    const float* w_lin  = (const float*)d_in[7];  // (100, 768)
    const float* b_lin  = (const float*)d_in[8];  // (100,)
    float*       out    = (float*)d_out;          // (128, 100)

    float* pun      = (float*)d_ws;                 // 128*768
    float* proj     = pun  + BB * EE;               // 128*2304  [q|k|v]
    float* ctx      = proj + BB * E3;               // 128*768
    float* attn_out = ctx  + BB * EE;               // 128*768

    // 1) gather + mean pooling
    pool_kernel<<<BB, 256, 0, stream>>>(lhs, loc, pun);

    // 2) in_proj: (128x768) @ (2304x768)^T + b -> (128x2304)
    gemm768_wmma<<<dim3(BB / 16, E3 / 16 / 8), 256, 0, stream>>>(
        pun, w_in, b_in, proj, E3);

    // 3) 16 heads of 128x128 attention (q scaled inside)
    attn_kernel<<<HH, 256, 0, stream>>>(proj, ctx);

    // 4) out_proj: (128x768) @ (768x768)^T + b -> (128x768)
    gemm768_wmma<<<dim3(BB / 16, EE / 16 / 8), 256, 0, stream>>>(
        ctx, w_out, b_out, attn_out, EE);

    // 5) final linear: (128x768) @ (100x768)^T + b -> (128x100)
    gemm768_wmma<<<dim3(BB / 16, 1), 256, 0, stream>>>(
        attn_out, w_lin, b_lin, out, SNN);
}